// HDConvEncoder_16716012716755
// MI455X (gfx1250) — compile-verified
//
#include <hip/hip_runtime.h>
#include <hip/hip_bf16.h>
#include <stdint.h>

// Problem constants (from reference): B=32, C=256, L=4096, S=8, Cs=32
#define L_LEN 4096
#define C_TOT 256
#define CS    32
#define S_GRP 8
#define TPB   256
#define EPT   (L_LEN / TPB)   // 16 elements per thread

// Issue async copies of one 16KB row (4096 f32) into LDS.
// Each thread covers bytes [tid*16 + r*4096, +16) for r=0..3.
// GVS mode: s[base] is the row's global base, VGPR holds the byte offset,
// VDST VGPR holds the LDS byte address (low 32 bits of the generic pointer
// to __shared__ == the workgroup-relative LDS offset).
__device__ __forceinline__ void async_row_load(const void* lds_dst,
                                               const float* gsrc, int tid) {
  uint32_t lbase = (uint32_t)(uintptr_t)lds_dst + (uint32_t)tid * 16u;
  uint32_t vbase = (uint32_t)tid * 16u;
  uint64_t sbase = (uint64_t)(uintptr_t)gsrc;
#pragma unroll
  for (int r = 0; r < 4; ++r) {
    uint32_t l = lbase + (uint32_t)r * 4096u;
    uint32_t v = vbase + (uint32_t)r * 4096u;
    asm volatile("global_load_async_to_lds_b128 %0, %1, %2"
                 :: "v"(l), "v"(v), "s"(sbase)
                 : "memory");
  }
}

__global__ __launch_bounds__(TPB) void hdconv_fused(
    const float* __restrict__ x,    // [B, C, L]
    const float* __restrict__ cw,   // [S, 3, Cs]
    const float* __restrict__ cb,   // [S, Cs]
    const float* __restrict__ lng,  // [L]
    const float* __restrict__ lnb,  // [L]
    const float* __restrict__ pww,  // [C]
    const float* __restrict__ pwb,  // [C]
    float* __restrict__ out)        // [B, C, L]
{
  __shared__ __align__(16) float xbuf0[L_LEN];
  __shared__ __align__(16) float xbuf1[L_LEN];
  __shared__ float tbuf[L_LEN + 2];   // halo-padded t ('same' conv padding)
  __shared__ float redS[8], redQ[8];
  __shared__ float stat[2];

  const int tid = threadIdx.x;
  const int b   = blockIdx.x >> 5;    // blockIdx = b*32 + j
  const int j   = blockIdx.x & 31;    // channel-in-group (chain id)
  const int wv  = tid >> 5;
  const int ln  = tid & 31;

  // ln_gamma/ln_beta positions repeat every group: preload into registers.
  float g[EPT], be[EPT];
#pragma unroll
  for (int r = 0; r < EPT; ++r) {
    g[r]  = lng[r * TPB + tid];
    be[r] = lnb[r * TPB + tid];
  }
  if (tid == 0) { tbuf[0] = 0.0f; tbuf[L_LEN + 1] = 0.0f; }

  // Prologue: start streaming group 0's row (channel c = j).
  const float* xrow0 = x + ((size_t)b * C_TOT + j) * L_LEN;
  async_row_load(xbuf0, xrow0, tid);

  float y[EPT];                       // recurrence carry y_{i-1}
#pragma unroll
  for (int r = 0; r < EPT; ++r) y[r] = 0.0f;

  for (int i = 0; i < S_GRP; ++i) {
    const float* xc = (i & 1) ? xbuf1 : xbuf0;
    float*       xn = (i & 1) ? xbuf0 : xbuf1;
    const int    c  = i * CS + j;

    // Pipeline: issue next row, then wait only for the current one
    // (async loads complete in order per wave: <=4 outstanding => cur done).
    if (i + 1 < S_GRP) {
      const float* xrow = x + ((size_t)b * C_TOT + (c + CS)) * L_LEN;
      async_row_load(xn, xrow, tid);
      asm volatile("s_wait_asynccnt 0x4" ::: "memory");
    } else {
      asm volatile("s_wait_asynccnt 0x0" ::: "memory");
    }
    __syncthreads();                  // current x row resident for all waves

    // t_i = x_i + y_{i-1}, staged with halo shift (tbuf[0]=tbuf[L+1]=0)
#pragma unroll
    for (int r = 0; r < EPT; ++r) {
      int l = r * TPB + tid;
      tbuf[l + 1] = xc[l] + y[r];
    }
    __syncthreads();                  // tbuf ready

    const float w0 = cw[i * 96 + j];
    const float w1 = cw[i * 96 + 32 + j];
    const float w2 = cw[i * 96 + 64 + j];
    const float bb = cb[i * 32 + j];

    // y_i = dwconv3(t_i); accumulate sum / sumsq for the LayerNorm over L.
    float s1 = 0.0f, s2 = 0.0f;
#pragma unroll
    for (int r = 0; r < EPT; ++r) {
      int l = r * TPB + tid;
      float v = fmaf(w0, tbuf[l], fmaf(w1, tbuf[l + 1], fmaf(w2, tbuf[l + 2], bb)));
      y[r] = v;
      s1 += v;
      s2 = fmaf(v, v, s2);
    }
#pragma unroll
    for (int off = 16; off >= 1; off >>= 1) {  // wave32 reduction
      s1 += __shfl_xor(s1, off, 32);
      s2 += __shfl_xor(s2, off, 32);
    }
    if (ln == 0) { redS[wv] = s1; redQ[wv] = s2; }
    __syncthreads();
    if (tid == 0) {
      float t1 = 0.0f, t2 = 0.0f;
#pragma unroll
      for (int w = 0; w < 8; ++w) { t1 += redS[w]; t2 += redQ[w]; }
      const float inv = 1.0f / (float)L_LEN;
      float mean = t1 * inv;
      float var  = fmaf(-mean, mean, t2 * inv);   // biased var (jnp.var)
      stat[0] = mean;
      stat[1] = rsqrtf(var + 1e-6f);
    }
    __syncthreads();
    const float mean = stat[0];
    const float rstd = stat[1];
    const float pwc  = pww[c];
    const float pbc  = pwb[c];

    // LN (gamma/beta indexed by l) -> per-channel scale/bias -> exact GELU
    // -> residual with the original input row (still resident in LDS).
    float* orow = out + ((size_t)b * C_TOT + c) * L_LEN;
#pragma unroll
    for (int r = 0; r < EPT; ++r) {
      int l = r * TPB + tid;
      float v = fmaf((y[r] - mean) * rstd, g[r], be[r]);
      v = fmaf(v, pwc, pbc);
      float ge = 0.5f * v * (1.0f + erff(v * 0.70710678118654752440f));
      orow[l] = xc[l] + ge;
    }
    __syncthreads();  // guard: xc becomes next iteration's async target; tbuf reused
  }
}

extern "C" void kernel_launch(void* const* d_in, const int* in_sizes, int n_in,
                              void* d_out, int out_size, void* d_ws, size_t ws_size,
                              hipStream_t stream) {
  const float* x   = (const float*)d_in[0];
  const float* cw  = (const float*)d_in[1];
  const float* cb  = (const float*)d_in[2];
  const float* lng = (const float*)d_in[3];
  const float* lnb = (const float*)d_in[4];
  const float* pww = (const float*)d_in[5];
  const float* pwb = (const float*)d_in[6];
  float* out = (float*)d_out;

  const int B = in_sizes[0] / (C_TOT * L_LEN);   // 32
  dim3 grid(B * CS);                             // one block per (b, j) chain
  dim3 block(TPB);
  hipLaunchKernelGGL(hdconv_fused, grid, block, 0, stream,
                     x, cw, cb, lng, lnb, pww, pwb, out);
}